// trainable_Euler_method_22239340659085
// MI455X (gfx1250) — compile-verified
//
#include <hip/hip_runtime.h>
#include <hip/hip_bf16.h>

typedef float v2f __attribute__((ext_vector_type(2)));
typedef float v8f __attribute__((ext_vector_type(8)));

#define HIDDEN     1024
#define LANES      32
#define PER_LANE   (HIDDEN / LANES)   // 32 floats per lane
#define PER_LANE2  (PER_LANE / 2)     // 16 float2 per lane

// Single-wave serial Euler integrator. All weights register-resident; the
// 32-lane dot-product reduction is two back-to-back V_WMMA_F32_16X16X4_F32
// ops (ones-matrix row sums), leaving the full sum uniform in every lane —
// no LDS, no shuffles, no barriers anywhere on the 4095-step chain.
__global__ __launch_bounds__(32)
void trainable_euler_kernel(const float* __restrict__ x0,
                            const float* __restrict__ dt,
                            const int*   __restrict__ ntp,
                            const float* __restrict__ W1,
                            const float* __restrict__ b1,
                            const float* __restrict__ W2,
                            const float* __restrict__ b2,
                            float* __restrict__ out)
{
    const int   lane = threadIdx.x;        // 0..31, one wave32
    const int   nt   = ntp[0];             // 4096
    const float dt_s = dt[0];
    const float b2_s = b2[0];
    float x = x0[0];

    // Each lane owns hidden units [lane*32, lane*32+32) -> contiguous b128 loads.
    v2f w1r[PER_LANE2], b1r[PER_LANE2], w2r[PER_LANE2];
    {
        const float4* W1v = (const float4*)(W1 + lane * PER_LANE);
        const float4* B1v = (const float4*)(b1 + lane * PER_LANE);
        const float4* W2v = (const float4*)(W2 + lane * PER_LANE);
#pragma unroll
        for (int j = 0; j < PER_LANE2 / 2; ++j) {
            float4 a = W1v[j], bb = B1v[j], c = W2v[j];
            w1r[2*j+0] = (v2f){a.x,  a.y};  w1r[2*j+1] = (v2f){a.z,  a.w};
            b1r[2*j+0] = (v2f){bb.x, bb.y}; b1r[2*j+1] = (v2f){bb.z, bb.w};
            w2r[2*j+0] = (v2f){c.x,  c.y};  w2r[2*j+1] = (v2f){c.z,  c.w};
        }
    }

    if (lane == 0) out[0] = x;             // x_pred[0] = x0

    const v2f ones = {1.0f, 1.0f};         // B = ones(4x16): WMMA does row sums

    for (int k = 1; k < nt; ++k) {
        // --- per-lane partial of relu(x*W1+b1) . W2 ---
        // 4 independent accumulators break the dependent-FMA chain (depth 4).
        v2f xx = {x, x};
        v2f a0 = {0.0f, 0.0f}, a1 = {0.0f, 0.0f};
        v2f a2 = {0.0f, 0.0f}, a3 = {0.0f, 0.0f};
#pragma unroll
        for (int j = 0; j < PER_LANE2; j += 4) {
            v2f h0 = xx * w1r[j+0] + b1r[j+0];
            v2f h1 = xx * w1r[j+1] + b1r[j+1];
            v2f h2 = xx * w1r[j+2] + b1r[j+2];
            v2f h3 = xx * w1r[j+3] + b1r[j+3];
            h0.x = fmaxf(h0.x, 0.0f); h0.y = fmaxf(h0.y, 0.0f);
            h1.x = fmaxf(h1.x, 0.0f); h1.y = fmaxf(h1.y, 0.0f);
            h2.x = fmaxf(h2.x, 0.0f); h2.y = fmaxf(h2.y, 0.0f);
            h3.x = fmaxf(h3.x, 0.0f); h3.y = fmaxf(h3.y, 0.0f);
            a0 = h0 * w2r[j+0] + a0;
            a1 = h1 * w2r[j+1] + a1;
            a2 = h2 * w2r[j+2] + a2;
            a3 = h3 * w2r[j+3] + a3;
        }
        v2f acc2 = (a0 + a1) + (a2 + a3);

        // --- WMMA #1: A(16x4) = acc2 (x -> K0/K2, y -> K1/K3), B = ones.
        // Row sum U_M = acc2.x[M]+acc2.y[M]+acc2.x[M+16]+acc2.y[M+16];
        // d[v]: lanes 0-15 = U_v, lanes 16-31 = U_{v+8} (replicated over N).
        v8f d = {};
        d = __builtin_amdgcn_wmma_f32_16x16x4_f32(false, acc2, false, ones,
                                                  (short)0, d, false, false);
        // Collapse 8 D regs -> one v2f on naturally adjacent pairs (no movs).
        v2f p = __builtin_shufflevector(d, d, 0, 1) +
                __builtin_shufflevector(d, d, 2, 3);
        v2f q = __builtin_shufflevector(d, d, 4, 5) +
                __builtin_shufflevector(d, d, 6, 7);
        v2f r = p + q;
        // r.x/r.y now hold even/odd partial sums per half-wave.

        // --- WMMA #2: A = r, B = ones. Every row sum = full 32-lane sum,
        // so d2[0] holds S uniformly in all lanes (replaces shfl + add).
        v8f d2 = {};
        d2 = __builtin_amdgcn_wmma_f32_16x16x4_f32(false, r, false, ones,
                                                   (short)0, d2, false, false);
        float s = d2[0];

        // --- uniform Euler update (no broadcast needed) ---
        x = fmaf(dt_s, s + b2_s, x);
        if (lane == 0) out[k] = x;
    }
}

extern "C" void kernel_launch(void* const* d_in, const int* in_sizes, int n_in,
                              void* d_out, int out_size, void* d_ws, size_t ws_size,
                              hipStream_t stream) {
    (void)in_sizes; (void)n_in; (void)out_size; (void)d_ws; (void)ws_size;
    const float* x0 = (const float*)d_in[0];
    const float* dt = (const float*)d_in[1];
    const int*   nt = (const int*)  d_in[2];
    const float* W1 = (const float*)d_in[3];
    const float* b1 = (const float*)d_in[4];
    const float* W2 = (const float*)d_in[5];
    const float* b2 = (const float*)d_in[6];
    float* out = (float*)d_out;

    // Inherently serial recurrence: one wave, one WGP, minimum-latency loop.
    trainable_euler_kernel<<<1, 32, 0, stream>>>(x0, dt, nt, W1, b1, W2, b2, out);
}